// PUnetStepLayered_23338852287253
// MI455X (gfx1250) — compile-verified
//
#include <hip/hip_runtime.h>
#include <hip/hip_bf16.h>

typedef float v2f __attribute__((ext_vector_type(2)));
typedef float v8f __attribute__((ext_vector_type(8)));

// ---------------- utility kernels ----------------

__global__ __launch_bounds__(256) void k_zero(float* p, size_t n) {
  size_t i = (size_t)blockIdx.x * blockDim.x + threadIdx.x;
  size_t stride = (size_t)gridDim.x * blockDim.x;
  for (; i < n; i += stride) p[i] = 0.0f;
}

__global__ __launch_bounds__(256) void k_scatter_edges(const int* __restrict__ ei,
                                                       float* __restrict__ A,
                                                       int E, int ld) {
  int e = blockIdx.x * blockDim.x + threadIdx.x;
  if (e >= E) return;
  int src = ei[e];      // edge_index[0, e]
  int dst = ei[E + e];  // edge_index[1, e]
  atomicAdd(&A[(size_t)dst * ld + src], 1.0f);
}

// dis[i] = 1/sqrt(rowsum(A[i,:]) + 2)   (Ah = A + 2I)
__global__ __launch_bounds__(128) void k_row_rsqrt(const float* __restrict__ A,
                                                   float* __restrict__ dis,
                                                   int n, int ld) {
  int i = blockIdx.x;
  int tid = threadIdx.x;
  float s = 0.0f;
  const float* row = A + (size_t)i * ld;
  for (int j = tid; j < n; j += 128) s += row[j];
  __shared__ float red[128];
  red[tid] = s;
  __syncthreads();
  for (int off = 64; off > 0; off >>= 1) {
    if (tid < off) red[tid] += red[tid + off];
    __syncthreads();
  }
  if (tid == 0) {
    float d = red[0] + 2.0f;
    dis[i] = (d > 0.0f) ? rsqrtf(d) : 0.0f;
  }
}

// An[i,j] = dis[i]*(A[i,j] + 2*(i==j))*dis[j], zero in pad region
__global__ __launch_bounds__(256) void k_build_An(const float* __restrict__ A,
                                                  const float* __restrict__ dis,
                                                  float* __restrict__ An,
                                                  int n, int np) {
  size_t idx = (size_t)blockIdx.x * blockDim.x + threadIdx.x;
  if (idx >= (size_t)np * np) return;
  int i = (int)(idx / np), j = (int)(idx % np);
  float v = 0.0f;
  if (i < n && j < n) {
    float a = A[(size_t)i * np + j];
    if (i == j) a += 2.0f;
    v = dis[i] * a * dis[j];
  }
  An[idx] = v;
}

__global__ __launch_bounds__(256) void k_set_diag1(float* __restrict__ A, int n, int ld) {
  int i = blockIdx.x * blockDim.x + threadIdx.x;
  if (i < n) A[(size_t)i * ld + i] = 1.0f;
}

// Y[m, :] = X[m, :] @ W  (Kc x Nc weights, row-major); zero pad rows
__global__ __launch_bounds__(256) void k_xw(const float* __restrict__ X,
                                            const float* __restrict__ W,
                                            float* __restrict__ Y,
                                            int Mreal, int Mpad, int Kc, int Nc, int ldx) {
  int idx = blockIdx.x * blockDim.x + threadIdx.x;
  if (idx >= Mpad * Nc) return;
  int m = idx / Nc, c = idx % Nc;
  float s = 0.0f;
  if (m < Mreal) {
    const float* xr = X + (size_t)m * ldx;
    for (int k = 0; k < Kc; ++k) s += xr[k] * W[k * Nc + c];
  }
  Y[(size_t)m * Nc + c] = s;
}

__global__ __launch_bounds__(128) void k_score(const float* __restrict__ H,
                                               const float* __restrict__ p,
                                               float* __restrict__ score,
                                               int n, int hc) {
  int i = blockIdx.x * blockDim.x + threadIdx.x;
  if (i >= n) return;
  float pn = 0.0f;
  for (int k = 0; k < hc; ++k) pn += p[k] * p[k];
  pn = sqrtf(pn);
  const float* hr = H + (size_t)i * hc;
  float s = 0.0f;
  for (int k = 0; k < hc; ++k) s += hr[k] * p[k];
  score[i] = tanhf(s / pn);
}

// exact top-k by rank; stable tie-break on lower index (matches lax.top_k)
__global__ __launch_bounds__(128) void k_topk(const float* __restrict__ score,
                                              int* __restrict__ perm, int n, int Ksel) {
  int i = blockIdx.x * blockDim.x + threadIdx.x;
  if (i >= n) return;
  float si = score[i];
  int rank = 0;
  for (int j = 0; j < n; ++j) {
    float sj = score[j];
    rank += (sj > si) || (sj == si && j < i);
  }
  if (rank < Ksel) perm[rank] = i;
}

__global__ __launch_bounds__(256) void k_gather_feat(const float* __restrict__ H,
                                                     const int* __restrict__ perm,
                                                     const float* __restrict__ score,
                                                     float* __restrict__ Hp,
                                                     int Ksel, int Kpad, int hc) {
  int idx = blockIdx.x * blockDim.x + threadIdx.x;
  if (idx >= Kpad * hc) return;
  int r = idx / hc, c = idx % hc;
  float v = 0.0f;
  if (r < Ksel) {
    int p = perm[r];
    v = H[(size_t)p * hc + c] * score[p];
  }
  Hp[idx] = v;
}

__global__ __launch_bounds__(256) void k_gather_adj(const float* __restrict__ Asrc,
                                                    const int* __restrict__ perm,
                                                    float* __restrict__ Adst,
                                                    int Ksel, int Kpad, int lds) {
  size_t idx = (size_t)blockIdx.x * blockDim.x + threadIdx.x;
  if (idx >= (size_t)Kpad * Kpad) return;
  int r = (int)(idx / Kpad), c = (int)(idx % Kpad);
  float v = 0.0f;
  if (r < Ksel && c < Ksel) v = Asrc[(size_t)perm[r] * lds + perm[c]];
  Adst[idx] = v;
}

__global__ __launch_bounds__(256) void k_copy_pad(const float* __restrict__ src,
                                                  float* __restrict__ dst,
                                                  int Mreal, int Mpad, int hc) {
  int idx = blockIdx.x * blockDim.x + threadIdx.x;
  if (idx >= Mpad * hc) return;
  int m = idx / hc;
  dst[idx] = (m < Mreal) ? src[idx] : 0.0f;
}

__global__ __launch_bounds__(256) void k_scatter_add(float* __restrict__ U,
                                                     const int* __restrict__ perm,
                                                     const float* __restrict__ Hs,
                                                     int Ksel, int hc) {
  int idx = blockIdx.x * blockDim.x + threadIdx.x;
  if (idx >= Ksel * hc) return;
  int r = idx / hc, c = idx % hc;
  U[(size_t)perm[r] * hc + c] += Hs[idx];
}

// out[i, 0..2] = An0[i,:n] @ UW[:, 0..2] + b + 0.1*z[i, 0..2]
__global__ __launch_bounds__(128) void k_final(const float* __restrict__ An0,
                                               const float* __restrict__ UW,
                                               const float* __restrict__ bup,
                                               const float* __restrict__ z,
                                               float* __restrict__ out,
                                               int n, int np) {
  int i = blockIdx.x;
  int tid = threadIdx.x;
  const float* row = An0 + (size_t)i * np;
  float a0 = 0.f, a1 = 0.f, a2 = 0.f;
  for (int k = tid; k < n; k += 128) {
    float a = row[k];
    a0 += a * UW[k * 3 + 0];
    a1 += a * UW[k * 3 + 1];
    a2 += a * UW[k * 3 + 2];
  }
  __shared__ float red[3][128];
  red[0][tid] = a0; red[1][tid] = a1; red[2][tid] = a2;
  __syncthreads();
  for (int off = 64; off > 0; off >>= 1) {
    if (tid < off) {
      red[0][tid] += red[0][tid + off];
      red[1][tid] += red[1][tid + off];
      red[2][tid] += red[2][tid + off];
    }
    __syncthreads();
  }
  if (tid < 3) out[i * 3 + tid] = red[tid][0] + bup[tid] + 0.1f * z[i * 3 + tid];
}

// ---------------- WMMA fp32 GEMM ----------------
// C[M,N] = A[M,K] @ B[K,N]; M, N multiples of 64, K multiple of 16
// (all buffers zero-padded to these sizes by their producers).
// Block: 128 threads = 4 waves; block tile 64x64; wave tile 16x64.
// Double-buffered (ping-pong) fragment loads: loads for K-chunk n+1 are
// issued before the WMMAs of chunk n, so s_wait_loadcnt overlaps MMA time.
// Epilogue: optional +bias[col], tanh, zero-diagonal (for augment()).

#define LOAD_FRAGS(a0, a1, b0, b1, kk)                                      \
  do {                                                                      \
    const float* Bk0 = Bbase + (size_t)(kk)*ldb;                            \
    const float* Bk1 = Bk0 + (size_t)4 * ldb;                               \
    (a0) = *(const v2f*)(Arow + (kk));                                      \
    (a1) = *(const v2f*)(Arow + (kk) + 4);                                  \
    _Pragma("unroll") for (int t = 0; t < 4; ++t) {                         \
      (b0)[t].x = Bk0[16 * t];                                              \
      (b0)[t].y = Bk0[16 * t + ldb];                                        \
    }                                                                       \
    _Pragma("unroll") for (int t = 0; t < 4; ++t) {                         \
      (b1)[t].x = Bk1[16 * t];                                              \
      (b1)[t].y = Bk1[16 * t + ldb];                                        \
    }                                                                       \
  } while (0)

#define MMA_FRAGS(a0, a1, b0, b1)                                           \
  do {                                                                      \
    _Pragma("unroll") for (int t = 0; t < 4; ++t)                           \
        acc[t] = __builtin_amdgcn_wmma_f32_16x16x4_f32(                     \
            false, (a0), false, (b0)[t], (short)0, acc[t], false, false);   \
    _Pragma("unroll") for (int t = 0; t < 4; ++t)                           \
        acc[t] = __builtin_amdgcn_wmma_f32_16x16x4_f32(                     \
            false, (a1), false, (b1)[t], (short)0, acc[t], false, false);   \
  } while (0)

__global__ __launch_bounds__(128)
void gemm_f32_wmma(const float* __restrict__ A, const float* __restrict__ B,
                   float* __restrict__ C,
                   int K, int lda, int ldb, int ldc,
                   const float* __restrict__ bias, int do_tanh, int zero_diag) {
  const int lane = threadIdx.x & 31;
  const int wave = threadIdx.x >> 5;
  const int m16  = lane & 15;
  const int kh   = lane >> 4;                 // lane half selects K pair {2kh, 2kh+1}
  const int rowBase = blockIdx.y * 64 + wave * 16;
  const int colBase = blockIdx.x * 64;

  // A fragment: 16x4 f32 tile, lane m16 = row, kh = K-pair
  const float* Arow  = A + (size_t)(rowBase + m16) * lda + 2 * kh;
  // B fragment: row (k + 2kh), column colBase + 16t + m16
  const float* Bbase = B + (size_t)(2 * kh) * ldb + colBase + m16;

  v8f acc[4] = {};

  v2f a0A, a1A, b0A[4], b1A[4];
  v2f a0B, a1B, b0B[4], b1B[4];

  LOAD_FRAGS(a0A, a1A, b0A, b1A, 0);
  int k = 0;
  for (; k + 16 < K; k += 16) {
    LOAD_FRAGS(a0B, a1B, b0B, b1B, k + 8);
    MMA_FRAGS(a0A, a1A, b0A, b1A);
    LOAD_FRAGS(a0A, a1A, b0A, b1A, k + 16);
    MMA_FRAGS(a0B, a1B, b0B, b1B);
  }
  // tail: chunks k and k+8 (K - k == 16)
  LOAD_FRAGS(a0B, a1B, b0B, b1B, k + 8);
  MMA_FRAGS(a0A, a1A, b0A, b1A);
  MMA_FRAGS(a0B, a1B, b0B, b1B);

#pragma unroll
  for (int t = 0; t < 4; ++t) {
    int col = colBase + 16 * t + m16;
    float bv = bias ? bias[col] : 0.0f;
#pragma unroll
    for (int r = 0; r < 8; ++r) {
      int row = rowBase + r + 8 * kh;         // C/D layout: lane half offsets M by 8
      float v = acc[t][r] + bv;
      if (do_tanh) v = tanhf(v);
      if (zero_diag && row == col) v = 0.0f;
      C[(size_t)row * ldc + col] = v;
    }
  }
}

// ---------------- launch ----------------

static inline int cdiv(int a, int b) { return (a + b - 1) / b; }

extern "C" void kernel_launch(void* const* d_in, const int* in_sizes, int n_in,
                              void* d_out, int out_size, void* d_ws, size_t ws_size,
                              hipStream_t stream) {
  (void)in_sizes; (void)n_in; (void)ws_size;
  const int Nn = 3000, Ee = 48000, INC = 32, HID = 128, OUTC = 3;
  const int K1 = 1500, K2 = 750;
  const int NP = 3008, K1P = 1536, K2P = 768;   // padded to multiples of 64

  const float* x   = (const float*)d_in[0];
  const float* z   = (const float*)d_in[1];
  const float* W0  = (const float*)d_in[2];
  const float* b0  = (const float*)d_in[3];
  const float* W1  = (const float*)d_in[4];
  const float* b1  = (const float*)d_in[5];
  const float* W2  = (const float*)d_in[6];
  const float* b2  = (const float*)d_in[7];
  const float* p1  = (const float*)d_in[8];
  const float* p2  = (const float*)d_in[9];
  const float* Wu0 = (const float*)d_in[10];
  const float* bu0 = (const float*)d_in[11];
  const float* Wu1 = (const float*)d_in[12];
  const float* bu1 = (const float*)d_in[13];
  const int*   ei  = (const int*)d_in[14];
  float* out = (float*)d_out;

  const size_t S_N  = (size_t)NP  * NP  * 4;
  const size_t S_K1 = (size_t)K1P * K1P * 4;
  const size_t S_K2 = (size_t)K2P * K2P * 4;
  char* w = (char*)d_ws;

  float* A0    = (float*)(w);              // A0 -> B0; region later reused
  float* An0   = (float*)(w + S_N);
  float* Aaug0 = (float*)(w + 2 * S_N);
  // reuse of region 0 after augment(A0) consumed B0:
  float* A1    = (float*)(w);
  float* An1   = (float*)(w + S_K1);
  float* Aaug1 = (float*)(w + 2 * S_K1);
  float* A2    = (float*)(w + 3 * S_K1);
  float* An2   = (float*)(w + 3 * S_K1 + S_K2);

  size_t o = 3 * S_N;
  auto take = [&](size_t bytes) { char* p = w + o; o += (bytes + 255) & ~(size_t)255; return p; };
  float* dis0 = (float*)take((size_t)NP * 4);
  float* dis1 = (float*)take((size_t)K1P * 4);
  float* dis2 = (float*)take((size_t)K2P * 4);
  float* sc1  = (float*)take((size_t)NP * 4);
  float* sc2  = (float*)take((size_t)K1P * 4);
  int*   perm1= (int*)  take((size_t)NP * 4);
  int*   perm2= (int*)  take((size_t)K1P * 4);
  float* XW   = (float*)take((size_t)NP  * HID * 4);
  float* h0   = (float*)take((size_t)NP  * HID * 4);
  float* hp1  = (float*)take((size_t)K1P * HID * 4);
  float* h1   = (float*)take((size_t)K1P * HID * 4);
  float* hp2  = (float*)take((size_t)K2P * HID * 4);
  float* h2   = (float*)take((size_t)K2P * HID * 4);
  float* u1   = (float*)take((size_t)K1P * HID * 4);
  float* u1b  = (float*)take((size_t)K1P * HID * 4);
  float* u0   = (float*)take((size_t)NP  * HID * 4);
  float* UW   = (float*)take((size_t)NP  * 4 * 4);

  // ---- build A0, An0, B0 ----
  k_zero<<<cdiv(NP * NP, 256), 256, 0, stream>>>(A0, (size_t)NP * NP);
  k_zero<<<cdiv(out_size, 256), 256, 0, stream>>>(out, (size_t)out_size);
  k_scatter_edges<<<cdiv(Ee, 256), 256, 0, stream>>>(ei, A0, Ee, NP);
  k_row_rsqrt<<<Nn, 128, 0, stream>>>(A0, dis0, Nn, NP);
  k_build_An<<<cdiv(NP * NP, 256), 256, 0, stream>>>(A0, dis0, An0, Nn, NP);
  k_set_diag1<<<cdiv(Nn, 256), 256, 0, stream>>>(A0, Nn, NP);   // A0 -> B0

  // ---- h0 = tanh(An0 @ (x @ W0) + b0) ----
  k_xw<<<cdiv(NP * HID, 256), 256, 0, stream>>>(x, W0, XW, Nn, NP, INC, HID, INC);
  gemm_f32_wmma<<<dim3(HID / 64, NP / 64), 128, 0, stream>>>(
      An0, XW, h0, NP, NP, HID, HID, b0, 1, 0);

  // ---- Aaug0 = zero_diag(B0 @ B0)  (the 54-GFLOP term) ----
  gemm_f32_wmma<<<dim3(NP / 64, NP / 64), 128, 0, stream>>>(
      A0, A0, Aaug0, NP, NP, NP, NP, nullptr, 0, 1);

  // ---- pool 1 ----
  k_score<<<cdiv(Nn, 128), 128, 0, stream>>>(h0, p1, sc1, Nn, HID);
  k_topk<<<cdiv(Nn, 128), 128, 0, stream>>>(sc1, perm1, Nn, K1);
  k_gather_feat<<<cdiv(K1P * HID, 256), 256, 0, stream>>>(h0, perm1, sc1, hp1, K1, K1P, HID);
  k_gather_adj<<<cdiv(K1P * K1P, 256), 256, 0, stream>>>(Aaug0, perm1, A1, K1, K1P, NP);

  // ---- h1 = tanh(An1 @ (hp1 @ W1) + b1) ----
  k_row_rsqrt<<<K1, 128, 0, stream>>>(A1, dis1, K1, K1P);
  k_build_An<<<cdiv(K1P * K1P, 256), 256, 0, stream>>>(A1, dis1, An1, K1, K1P);
  k_set_diag1<<<cdiv(K1, 256), 256, 0, stream>>>(A1, K1, K1P);  // A1 -> B1
  k_xw<<<cdiv(K1P * HID, 256), 256, 0, stream>>>(hp1, W1, XW, K1, K1P, HID, HID, HID);
  gemm_f32_wmma<<<dim3(HID / 64, K1P / 64), 128, 0, stream>>>(
      An1, XW, h1, K1P, K1P, HID, HID, b1, 1, 0);

  // ---- Aaug1 = zero_diag(B1 @ B1) ----
  gemm_f32_wmma<<<dim3(K1P / 64, K1P / 64), 128, 0, stream>>>(
      A1, A1, Aaug1, K1P, K1P, K1P, K1P, nullptr, 0, 1);

  // ---- pool 2 ----
  k_score<<<cdiv(K1, 128), 128, 0, stream>>>(h1, p2, sc2, K1, HID);
  k_topk<<<cdiv(K1, 128), 128, 0, stream>>>(sc2, perm2, K1, K2);
  k_gather_feat<<<cdiv(K2P * HID, 256), 256, 0, stream>>>(h1, perm2, sc2, hp2, K2, K2P, HID);
  k_gather_adj<<<cdiv(K2P * K2P, 256), 256, 0, stream>>>(Aaug1, perm2, A2, K2, K2P, K1P);

  // ---- h2 = tanh(An2 @ (hp2 @ W2) + b2) ----
  k_row_rsqrt<<<K2, 128, 0, stream>>>(A2, dis2, K2, K2P);
  k_build_An<<<cdiv(K2P * K2P, 256), 256, 0, stream>>>(A2, dis2, An2, K2, K2P);
  k_xw<<<cdiv(K2P * HID, 256), 256, 0, stream>>>(hp2, W2, XW, K2, K2P, HID, HID, HID);
  gemm_f32_wmma<<<dim3(HID / 64, K2P / 64), 128, 0, stream>>>(
      An2, XW, h2, K2P, K2P, HID, HID, b2, 1, 0);

  // ---- up path ----
  k_copy_pad<<<cdiv(K1P * HID, 256), 256, 0, stream>>>(h1, u1, K1, K1P, HID);
  k_scatter_add<<<cdiv(K2 * HID, 256), 256, 0, stream>>>(u1, perm2, h2, K2, HID);
  k_xw<<<cdiv(K1P * HID, 256), 256, 0, stream>>>(u1, Wu0, XW, K1, K1P, HID, HID, HID);
  gemm_f32_wmma<<<dim3(HID / 64, K1P / 64), 128, 0, stream>>>(
      An1, XW, u1b, K1P, K1P, HID, HID, bu0, 1, 0);

  k_copy_pad<<<cdiv(NP * HID, 256), 256, 0, stream>>>(h0, u0, Nn, NP, HID);
  k_scatter_add<<<cdiv(K1 * HID, 256), 256, 0, stream>>>(u0, perm1, u1b, K1, HID);

  // ---- final GCN (HID -> 3) + noise; label part of d_out stays zero ----
  k_xw<<<cdiv(Nn * OUTC, 256), 256, 0, stream>>>(u0, Wu1, UW, Nn, Nn, HID, OUTC, HID);
  k_final<<<Nn, 128, 0, stream>>>(An0, UW, bu1, z, out, Nn, NP);
}